// RoiPoolingLayer_81767587381900
// MI455X (gfx1250) — compile-verified
//
#include <hip/hip_runtime.h>

typedef __attribute__((ext_vector_type(2))) float v2f;
typedef __attribute__((ext_vector_type(8))) float v8f;

#define POOL   14
#define KTOT   56          // 4 diagonal blocks x 14 taps
#define HW     128
#define CH     1024
#define NTILE  (CH / 16)   // 64 channel tiles of 16

__global__ __launch_bounds__(128) void RoiPool_wmma_kernel(
    const float* __restrict__ feat,   // (128,128,1024) NHWC, single image
    const float* __restrict__ rois,   // (N,4) x,y,w,h
    float* __restrict__ out,          // (N,14,14,1024)
    int nroi)
{
    __shared__ float sW[16 * KTOT];   // A-matrix weights, [px][k], rows 14,15 zero
    __shared__ int   sTap[KTOT];      // feature-element base offset of each tap

    const int tid   = threadIdx.x;
    const int block = blockIdx.x;
    const int r     = block / POOL;
    const int py    = block - r * POOL;
    if (r >= nroi) return;

    // ---- zero the weight matrix ----
    for (int i = tid; i < 16 * KTOT; i += blockDim.x) sW[i] = 0.0f;
    __syncthreads();

    // ---- threads 0..13: one output column each; fill taps + weights ----
    if (tid < POOL) {
        const float x = rois[r * 4 + 0];
        const float y = rois[r * 4 + 1];
        const float w = rois[r * 4 + 2];
        const float h = rois[r * 4 + 3];
        const float INV_STRIDE = 1.0f / 16.0f;

        const int xs = (int)rintf((x - 0.5f * w) * INV_STRIDE);  // x-axis start ("r")
        const int ys = (int)rintf((y - 0.5f * h) * INV_STRIDE);  // y-axis start ("c")
        const int wf = max((int)rintf(w * INV_STRIDE), 1);
        const int hf = max((int)rintf(h * INV_STRIDE), 1);

        // y axis sample for this py (TF half-pixel, unclamped frac, clamped idx)
        const float scy  = (float)hf / (float)POOL;
        const float srcy = ((float)py + 0.5f) * scy - 0.5f;
        const float fly  = floorf(srcy);
        const float ty   = srcy - fly;
        const float hiy  = (float)(hf - 1);
        int y0 = (int)fminf(fmaxf(fly,        0.0f), hiy) + ys;
        int y1 = (int)fminf(fmaxf(fly + 1.0f, 0.0f), hiy) + ys;
        y0 = min(max(y0, 0), HW - 1);
        y1 = min(max(y1, 0), HW - 1);

        // x axis sample for px = tid
        const float scx  = (float)wf / (float)POOL;
        const float srcx = ((float)tid + 0.5f) * scx - 0.5f;
        const float flx  = floorf(srcx);
        const float tx   = srcx - flx;
        const float hix  = (float)(wf - 1);
        int x0 = (int)fminf(fmaxf(flx,        0.0f), hix) + xs;
        int x1 = (int)fminf(fmaxf(flx + 1.0f, 0.0f), hix) + xs;
        x0 = min(max(x0, 0), HW - 1);
        x1 = min(max(x1, 0), HW - 1);

        sTap[tid]      = (y0 * HW + x0) * CH;
        sTap[14 + tid] = (y0 * HW + x1) * CH;
        sTap[28 + tid] = (y1 * HW + x0) * CH;
        sTap[42 + tid] = (y1 * HW + x1) * CH;

        // A[px=tid][k]: diagonal-block bilinear weights (exact factorization)
        sW[tid * KTOT + tid]      = (1.0f - ty) * (1.0f - tx);
        sW[tid * KTOT + 14 + tid] = (1.0f - ty) * tx;
        sW[tid * KTOT + 28 + tid] = ty * (1.0f - tx);
        sW[tid * KTOT + 42 + tid] = ty * tx;
    }
    __syncthreads();

    // ---- per-wave WMMA loop over channel tiles ----
    const int lane    = tid & 31;
    const int wave    = tid >> 5;
    const int nwave   = blockDim.x >> 5;
    const int lc      = lane & 15;                 // N index (channel within tile)
    const int koff    = (lane < 16) ? 0 : 2;       // K sub-pair per half-wave
    const int px_base = (lane < 16) ? 0 : 8;       // D-row split per half-wave

    // A fragments: identical for every channel tile -> load once into VGPRs
    v2f a[14];
#pragma unroll
    for (int s = 0; s < 14; ++s) {
        const float* wp = &sW[lc * KTOT + 4 * s + koff];
        a[s].x = wp[0];
        a[s].y = wp[1];
    }
    // per-lane tap bases for its two K slots per step
    int tapA[14], tapB[14];
#pragma unroll
    for (int s = 0; s < 14; ++s) {
        tapA[s] = sTap[4 * s + koff];
        tapB[s] = sTap[4 * s + koff + 1];
    }

    const size_t outBase = (size_t)block * (size_t)(POOL * CH); // ((r*14+py)*14+px)*CH

    for (int cb = wave; cb < NTILE; cb += nwave) {
        const int cbase = cb * 16 + lc;
        v8f acc = {0.f, 0.f, 0.f, 0.f, 0.f, 0.f, 0.f, 0.f};
#pragma unroll
        for (int s = 0; s < 14; ++s) {
            v2f b;
            b.x = feat[tapA[s] + cbase];   // 64B-contiguous per half-wave
            b.y = feat[tapB[s] + cbase];
            acc = __builtin_amdgcn_wmma_f32_16x16x4_f32(
                false, a[s], false, b, (short)0, acc, false, false);
        }
        // D VGPR j: px = j (lanes 0-15) / j+8 (lanes 16-31); N = channel
#pragma unroll
        for (int j = 0; j < 8; ++j) {
            const int px = px_base + j;
            if (px < POOL)
                out[outBase + (size_t)px * CH + cbase] = acc[j];
        }
    }
}

extern "C" void kernel_launch(void* const* d_in, const int* in_sizes, int n_in,
                              void* d_out, int out_size, void* d_ws, size_t ws_size,
                              hipStream_t stream) {
    const float* feat = (const float*)d_in[0];   // (1,128,128,1024) f32
    const float* rois = (const float*)d_in[1];   // (N,4) f32
    float* out = (float*)d_out;                  // (1,N,14,14,1024) f32
    const int nroi = in_sizes[1] / 4;

    dim3 grid(nroi * POOL);
    dim3 blk(128);
    RoiPool_wmma_kernel<<<grid, blk, 0, stream>>>(feat, rois, out, nroi);
}